// SparsifyWrapper_19198503813176
// MI455X (gfx1250) — compile-verified
//
#include <hip/hip_runtime.h>

// ---------------------------------------------------------------------------
// TopK Sparse Autoencoder forward, fused: encode GEMM (f32 WMMA) + streaming
// exact top-64 + sparse decode.  One workgroup owns 16 token rows and sweeps
// all 65536 latent columns.  Each wave computes TWO 16x16 tiles per chunk so
// every A fragment (ds_load_b64) feeds two v_wmma_f32_16x16x4_f32.
// ---------------------------------------------------------------------------

typedef float v2f __attribute__((ext_vector_type(2)));
typedef float v8f __attribute__((ext_vector_type(8)));

#define D_IN        768
#define D_SAE       65536
#define ROWS_PER_WG 16
#define TOPK        64
#define CAND_MAX    256          // exact worst case: 256 cols per chunk per row
#define CHUNK_COLS  256          // 8 waves x 2 tiles x 16 cols
#define NUM_CHUNKS  (D_SAE / CHUNK_COLS)
#define BLOCK_THREADS 256

// dynamic LDS partition sizes (floats/ints)
#define SM_XTILE   (ROWS_PER_WG * D_IN)        // 12288 f
#define SM_TOPV    (ROWS_PER_WG * TOPK)        // 1024 f
#define SM_TOPI    (ROWS_PER_WG * TOPK)        // 1024 i
#define SM_CANDV   (ROWS_PER_WG * CAND_MAX)    // 4096 f
#define SM_CANDI   (ROWS_PER_WG * CAND_MAX)    // 4096 i
#define SMEM_BYTES ((SM_XTILE + SM_TOPV + SM_TOPI + SM_CANDV + SM_CANDI + 3*ROWS_PER_WG) * 4)

__global__ __launch_bounds__(BLOCK_THREADS)
void sae_topk_fused(const float* __restrict__ x,
                    const float* __restrict__ Wenc,
                    const float* __restrict__ benc,
                    const float* __restrict__ Wdec,
                    const float* __restrict__ bdec,
                    float* __restrict__ out)
{
    extern __shared__ char smemRaw[];
    float* xTile   = (float*)smemRaw;                        // [16][768] (x - b_dec)
    float* topV    = xTile  + SM_XTILE;                      // [16][64]
    int*   topI    = (int*)(topV + SM_TOPV);                 // [16][64]
    float* candV   = (float*)(topI + SM_TOPI);               // [16][256]
    int*   candI   = (int*)(candV + SM_CANDV);               // [16][256]
    float* rowMin  = (float*)(candI + SM_CANDI);             // [16]
    int*   minPos  = (int*)(rowMin + ROWS_PER_WG);           // [16]
    int*   candCnt = minPos + ROWS_PER_WG;                   // [16]

    const int tid  = threadIdx.x;
    const int lane = tid & 31;
    const int wave = tid >> 5;
    const int h    = lane >> 4;      // half-wave select (K pair / M+8)
    const int m    = lane & 15;      // row (A) / col (B,C,D) within tile

    const int row0 = blockIdx.x * ROWS_PER_WG;

    // ---- stage (x - b_dec) tile into LDS ----
    for (int i = tid; i < ROWS_PER_WG * D_IN; i += BLOCK_THREADS) {
        int r = i / D_IN;
        int c = i - r * D_IN;
        xTile[i] = x[(size_t)(row0 + r) * D_IN + c] - bdec[c];
    }
    if (tid < ROWS_PER_WG) { rowMin[tid] = 0.0f; minPos[tid] = 0; candCnt[tid] = 0; }
    for (int i = tid; i < ROWS_PER_WG * TOPK; i += BLOCK_THREADS) { topV[i] = 0.0f; topI[i] = 0; }
    __syncthreads();

    // A fragment source: lane m holds row m; VGPR v / half h -> K = 2h + v
    const float* aPtr = xTile + m * D_IN + 2 * h;

    // ---- encode sweep: 256 chunks of 256 columns; two 16x16 WMMA tiles/wave ----
    for (int chunk = 0; chunk < NUM_CHUNKS; ++chunk) {
        const int col0 = chunk * CHUNK_COLS + wave * 32 + m;  // first tile's N lane
        // B fragment source: lane m holds col; VGPR v / half h -> K = 2h + v
        const float* bPtr = Wenc + (size_t)(2 * h) * D_SAE + col0;

        v8f acc0 = {};
        v8f acc1 = {};
#pragma unroll 4
        for (int k = 0; k < D_IN; k += 4) {
            v2f a = *(const v2f*)(aPtr + k);                  // ds_load_b64, shared by both tiles
            const size_t o0 = (size_t)k * D_SAE;              // row k+2h
            const size_t o1 = o0 + D_SAE;                     // row k+2h+1
            v2f b0, b1;
            b0.x = bPtr[o0];        b0.y = bPtr[o1];
            b1.x = bPtr[o0 + 16];   b1.y = bPtr[o1 + 16];     // +64B folds into inst offset
            acc0 = __builtin_amdgcn_wmma_f32_16x16x4_f32(
                false, a, false, b0, (short)0, acc0, false, false);
            acc1 = __builtin_amdgcn_wmma_f32_16x16x4_f32(
                false, a, false, b1, (short)0, acc1, false, false);
        }
        // prefetch next chunk's B stream head
        if (chunk + 1 < NUM_CHUNKS)
            __builtin_prefetch(bPtr + CHUNK_COLS, 0, 1);      // global_prefetch_b8

        // ---- epilogue: bias + relu + candidate filtering (both tiles) ----
#pragma unroll
        for (int t = 0; t < 2; ++t) {
            const int  col  = col0 + 16 * t;
            const float bias = benc[col];
            const v8f  acc  = t ? acc1 : acc0;
#pragma unroll
            for (int r = 0; r < 8; ++r) {
                float pre = acc[r] + bias;
                pre = fmaxf(pre, 0.0f);
                const int row = r + 8 * h;                    // C/D layout: M = vgpr + 8h
                if (pre > rowMin[row]) {                      // monotone threshold, stale-safe
                    int p = atomicAdd(&candCnt[row], 1);
                    if (p < CAND_MAX) {
                        candV[row * CAND_MAX + p] = pre;
                        candI[row * CAND_MAX + p] = col;
                    }
                }
            }
        }
        __syncthreads();

        // ---- per-row serial merge into exact top-64 ----
        if (tid < ROWS_PER_WG) {
            const int row = tid;
            int n = candCnt[row]; if (n > CAND_MAX) n = CAND_MAX;
            float  mn = rowMin[row];
            int    mp = minPos[row];
            float* tv = topV + row * TOPK;
            int*   ti = topI + row * TOPK;
            for (int p = 0; p < n; ++p) {
                float v = candV[row * CAND_MAX + p];
                if (v > mn) {
                    tv[mp] = v; ti[mp] = candI[row * CAND_MAX + p];
                    mn = tv[0]; mp = 0;
                    for (int q = 1; q < TOPK; ++q)
                        if (tv[q] < mn) { mn = tv[q]; mp = q; }
                }
            }
            rowMin[row] = mn; minPos[row] = mp; candCnt[row] = 0;
        }
        __syncthreads();
    }

    // ---- sparse decode: out = b_dec + sum_j v_j * W_dec[idx_j, :] ----
    const int drow = tid >> 4;          // 0..15 token row in tile
    const int dl   = tid & 15;          // 16 threads cover 768 cols as 192 float4
    float4 accv[12];
    const float4* bdec4 = (const float4*)bdec;
#pragma unroll
    for (int q = 0; q < 12; ++q) accv[q] = bdec4[dl + 16 * q];

    const float* tv = topV + drow * TOPK;
    const int*   ti = topI + drow * TOPK;
    for (int j = 0; j < TOPK; ++j) {
        const float v = tv[j];
        if (v <= 0.0f) continue;        // unfilled slots contribute nothing
        const float4* wrow = (const float4*)(Wdec + (size_t)ti[j] * D_IN);
#pragma unroll
        for (int q = 0; q < 12; ++q) {
            float4 w = wrow[dl + 16 * q];
            accv[q].x += v * w.x; accv[q].y += v * w.y;
            accv[q].z += v * w.z; accv[q].w += v * w.w;
        }
    }
    float4* out4 = (float4*)(out + (size_t)(row0 + drow) * D_IN);
#pragma unroll
    for (int q = 0; q < 12; ++q) out4[dl + 16 * q] = accv[q];
}

extern "C" void kernel_launch(void* const* d_in, const int* in_sizes, int n_in,
                              void* d_out, int out_size, void* d_ws, size_t ws_size,
                              hipStream_t stream)
{
    (void)in_sizes; (void)n_in; (void)d_ws; (void)ws_size; (void)out_size;
    const float* x    = (const float*)d_in[0];
    const float* Wenc = (const float*)d_in[1];
    const float* benc = (const float*)d_in[2];
    const float* Wdec = (const float*)d_in[3];
    const float* bdec = (const float*)d_in[4];
    float* out = (float*)d_out;

    static_assert(SMEM_BYTES < 320 * 1024, "LDS budget");
    (void)hipFuncSetAttribute((const void*)sae_topk_fused,
                              hipFuncAttributeMaxDynamicSharedMemorySize,
                              SMEM_BYTES);

    dim3 grid(2048 / ROWS_PER_WG);   // 128 workgroups, 16 token-rows each
    dim3 block(BLOCK_THREADS);
    sae_topk_fused<<<grid, block, SMEM_BYTES, stream>>>(x, Wenc, benc, Wdec, bdec, out);
}